// HGATLayer_49520972923352
// MI455X (gfx1250) — compile-verified
//
#include <hip/hip_runtime.h>
#include <hip/hip_bf16.h>

typedef __attribute__((ext_vector_type(2))) float v2f;
typedef __attribute__((ext_vector_type(8))) float v8f;

#define B_ 32
#define T_ 3
#define L_ 256
#define N_ 768
#define DIN_ 256
#define DOUT_ 128
#define ALPHA_ 0.2f
#define NEG_ -9.0e15f

// ---------------------------------------------------------------------------
// K1: Wh[b, t*L+l, e] = sum_d h[b, t*L+l, d] * W[t, d, e]
// One wave = one 16x16 f32 tile via V_WMMA_F32_16X16X4_F32.
// waves = B*T * (L/16) * (DOUT/16) = 96*16*8 = 12288
// ---------------------------------------------------------------------------
__global__ void __launch_bounds__(256) k_wh_gemm(const float* __restrict__ h,
                                                 const float* __restrict__ W,
                                                 float* __restrict__ Wh) {
  const int wave = blockIdx.x * 8 + (threadIdx.x >> 5);
  const int lane = threadIdx.x & 31;
  const int nt = wave & 7;
  const int mt = (wave >> 3) & 15;
  const int bt = wave >> 7;               // b*T + t, 0..95
  const int t = bt % T_;

  const float* __restrict__ A  = h + (size_t)bt * L_ * DIN_;     // [L, DIN]
  const float* __restrict__ Bw = W + (size_t)t * DIN_ * DOUT_;   // [DIN, DOUT]

  const int half = lane >> 4;             // 0: K+0..1, 1: K+2..3
  const int lr   = lane & 15;
  const int m = mt * 16 + lr;             // A row for this lane
  const int n = nt * 16 + lr;             // B col for this lane

  v8f acc = {};
  #pragma unroll 4
  for (int k0 = 0; k0 < DIN_; k0 += 4) {
    const int kk = k0 + half * 2;
    v2f a, b;
    a.x = A[(size_t)m * DIN_ + kk];
    a.y = A[(size_t)m * DIN_ + kk + 1];
    b.x = Bw[(size_t)kk * DOUT_ + n];
    b.y = Bw[(size_t)(kk + 1) * DOUT_ + n];
    acc = __builtin_amdgcn_wmma_f32_16x16x4_f32(false, a, false, b,
                                                (short)0, acc, false, false);
  }
  float* __restrict__ C = Wh + ((size_t)bt * L_ + mt * 16) * DOUT_ + nt * 16;
  #pragma unroll
  for (int v = 0; v < 8; ++v) {
    const int row = half * 8 + v;
    C[(size_t)row * DOUT_ + lr] = acc[v];
  }
}

// ---------------------------------------------------------------------------
// K2: w1[b,m,n,l] = Wh[b, m*L+l, :] . a_type[m,n,:128]
//     w2[b,m,n,l] = Wh[b, n*L+l, :] . a_type[m,n,128:]
// threads = B*T*T*L = 73728
// ---------------------------------------------------------------------------
__global__ void __launch_bounds__(256) k_w12(const float* __restrict__ Wh,
                                             const float* __restrict__ a_type,
                                             float* __restrict__ w1,
                                             float* __restrict__ w2) {
  const int idx = blockIdx.x * blockDim.x + threadIdx.x;
  if (idx >= B_ * T_ * T_ * L_) return;
  const int l = idx & (L_ - 1);
  const int mn = idx >> 8;
  const int n = mn % T_;
  const int m = (mn / T_) % T_;
  const int b = mn / (T_ * T_);

  const float4* __restrict__ at  = (const float4*)(a_type + (size_t)(m * T_ + n) * 2 * DOUT_);
  const float4* __restrict__ whm = (const float4*)(Wh + ((size_t)b * N_ + m * L_ + l) * DOUT_);
  const float4* __restrict__ whn = (const float4*)(Wh + ((size_t)b * N_ + n * L_ + l) * DOUT_);

  float acc1 = 0.f, acc2 = 0.f;
  #pragma unroll 8
  for (int e = 0; e < DOUT_ / 4; ++e) {
    float4 a1 = at[e];
    float4 a2 = at[e + DOUT_ / 4];
    float4 x1 = whm[e];
    float4 x2 = whn[e];
    acc1 += a1.x * x1.x + a1.y * x1.y + a1.z * x1.z + a1.w * x1.w;
    acc2 += a2.x * x2.x + a2.y * x2.y + a2.z * x2.z + a2.w * x2.w;
  }
  w1[idx] = acc1;
  w2[idx] = acc2;
}

// ---------------------------------------------------------------------------
// K3: e_type[b,m,n,l] = lrelu(w1[b,m,n,l] +
//                             sum_{l'} adj[b, m*L+l, n*L+l'] * w2[b,m,n,l'])
// One wave per (b,m,n,l) row; 73728 waves, wave32 shuffle reduce.
// ---------------------------------------------------------------------------
__global__ void __launch_bounds__(256) k_etype(const int* __restrict__ adj,
                                               const float* __restrict__ w1,
                                               const float* __restrict__ w2,
                                               float* __restrict__ e_type) {
  const int wave = blockIdx.x * 8 + (threadIdx.x >> 5);
  const int lane = threadIdx.x & 31;
  const int l = wave & (L_ - 1);
  const int mn = wave >> 8;
  const int n = mn % T_;
  const int m = (mn / T_) % T_;
  const int b = mn / (T_ * T_);

  const int* __restrict__ arow =
      adj + ((size_t)b * N_ + m * L_ + l) * N_ + n * L_;
  const float* __restrict__ w2row = w2 + ((size_t)mn) * L_;

  float s = 0.f;
  #pragma unroll
  for (int lp = lane; lp < L_; lp += 32)
    s += (float)arow[lp] * w2row[lp];

  #pragma unroll
  for (int off = 16; off > 0; off >>= 1)
    s += __shfl_xor(s, off, 32);

  if (lane == 0) {
    const float e = w1[wave] + s;
    e_type[wave] = e > 0.f ? e : ALPHA_ * e;
  }
}

// ---------------------------------------------------------------------------
// K4: node_type_att[b, m*L+l, n] = softmax_n(e_type[b,m,n,l])  (T_=3)
// threads = B*T*L = 24576; writes directly to output region 1.
// ---------------------------------------------------------------------------
__global__ void __launch_bounds__(256) k_typeatt(const float* __restrict__ e_type,
                                                 float* __restrict__ out_nta) {
  const int idx = blockIdx.x * blockDim.x + threadIdx.x;
  if (idx >= B_ * T_ * L_) return;
  const int l = idx & (L_ - 1);
  const int m = (idx >> 8) % T_;
  const int b = idx / (T_ * L_);

  const size_t base = ((size_t)b * T_ + m) * T_;
  const float e0 = e_type[(base + 0) * L_ + l];
  const float e1 = e_type[(base + 1) * L_ + l];
  const float e2 = e_type[(base + 2) * L_ + l];
  const float mx = fmaxf(e0, fmaxf(e1, e2));
  const float x0 = __expf(e0 - mx);
  const float x1 = __expf(e1 - mx);
  const float x2 = __expf(e2 - mx);
  const float inv = 1.f / (x0 + x1 + x2);
  float* __restrict__ o = out_nta + ((size_t)b * N_ + m * L_ + l) * T_;
  o[0] = x0 * inv;
  o[1] = x1 * inv;
  o[2] = x2 * inv;
}

// ---------------------------------------------------------------------------
// K5: s2[b,o,n'] = Wh[b,n',:] . a_node[o,128:]   (threads = B*T*N)
//     s1[b,n']   = Wh[b,n',:] . a_node[type(n'),:128]  (done by o==0 threads)
// ---------------------------------------------------------------------------
__global__ void __launch_bounds__(256) k_s12(const float* __restrict__ Wh,
                                             const float* __restrict__ a_node,
                                             float* __restrict__ s1,
                                             float* __restrict__ s2) {
  const int idx = blockIdx.x * blockDim.x + threadIdx.x;
  if (idx >= B_ * T_ * N_) return;
  const int np = idx % N_;
  const int o = (idx / N_) % T_;
  const int b = idx / (T_ * N_);

  const float4* __restrict__ wr = (const float4*)(Wh + ((size_t)b * N_ + np) * DOUT_);
  const float4* __restrict__ a2 = (const float4*)(a_node + (size_t)o * 2 * DOUT_ + DOUT_);

  float acc = 0.f;
  #pragma unroll 8
  for (int e = 0; e < DOUT_ / 4; ++e) {
    float4 a = a2[e];
    float4 x = wr[e];
    acc += a.x * x.x + a.y * x.y + a.z * x.z + a.w * x.w;
  }
  s2[idx] = acc;

  if (o == 0) {
    const int ot = np / L_;
    const float4* __restrict__ a1 = (const float4*)(a_node + (size_t)ot * 2 * DOUT_);
    float acc1 = 0.f;
    #pragma unroll 8
    for (int e = 0; e < DOUT_ / 4; ++e) {
      float4 a = a1[e];
      float4 x = wr[e];
      acc1 += a.x * x.x + a.y * x.y + a.z * x.z + a.w * x.w;
    }
    s1[(size_t)b * N_ + np] = acc1;
  }
}

// ---------------------------------------------------------------------------
// K6: one block per attention row (b, n). Computes both masked softmaxes
// (typed "attention" and "original_attention") and the regular rows needed
// by the final GEMM (only rows n < L).
// e_nn[n,n'] = lrelu(s1[b,n] + s2[b, type(n), n'])
// typed      = e_nn * node_type_att[b, n, type(n')]
// ---------------------------------------------------------------------------
__global__ void __launch_bounds__(256) k_attn_rows(const int* __restrict__ adj,
                                                   const float* __restrict__ s1,
                                                   const float* __restrict__ s2,
                                                   const float* __restrict__ nta,
                                                   float* __restrict__ out_att,
                                                   float* __restrict__ out_orig,
                                                   float* __restrict__ regular) {
  __shared__ float le[N_];
  __shared__ float lt[N_];
  __shared__ float rA[256];
  __shared__ float rB[256];

  const int row = blockIdx.x;             // b*N + n
  const int b = row / N_;
  const int n = row % N_;
  const int o = n / L_;
  const int tid = threadIdx.x;

  const float s1v = s1[row];
  const float ta0 = nta[(size_t)row * T_ + 0];
  const float ta1 = nta[(size_t)row * T_ + 1];
  const float ta2 = nta[(size_t)row * T_ + 2];

  const int* __restrict__ arow = adj + (size_t)row * N_;
  const float* __restrict__ s2row = s2 + ((size_t)b * T_ + o) * N_;

  float me = -3.0e38f, mt = -3.0e38f;
  #pragma unroll
  for (int p = 0; p < 3; ++p) {
    const int j = tid + p * 256;
    const int a = arow[j];
    float e = s1v + s2row[j];
    e = e > 0.f ? e : ALPHA_ * e;
    const float ta = (p == 0) ? ta0 : (p == 1 ? ta1 : ta2);
    const float t = e * ta;
    const float eM = a > 0 ? e : NEG_;
    const float tM = a > 0 ? t : NEG_;
    le[j] = eM;
    lt[j] = tM;
    me = fmaxf(me, eM);
    mt = fmaxf(mt, tM);
  }
  rA[tid] = me;
  rB[tid] = mt;
  __syncthreads();
  for (int s = 128; s > 0; s >>= 1) {
    if (tid < s) {
      rA[tid] = fmaxf(rA[tid], rA[tid + s]);
      rB[tid] = fmaxf(rB[tid], rB[tid + s]);
    }
    __syncthreads();
  }
  me = rA[0];
  mt = rB[0];
  __syncthreads();

  float se = 0.f, st = 0.f;
  #pragma unroll
  for (int p = 0; p < 3; ++p) {
    const int j = tid + p * 256;
    const float ee = __expf(le[j] - me);
    const float tt = __expf(lt[j] - mt);
    le[j] = ee;
    lt[j] = tt;
    se += ee;
    st += tt;
  }
  rA[tid] = se;
  rB[tid] = st;
  __syncthreads();
  for (int s = 128; s > 0; s >>= 1) {
    if (tid < s) {
      rA[tid] += rA[tid + s];
      rB[tid] += rB[tid + s];
    }
    __syncthreads();
  }
  const float ise = 1.f / rA[0];
  const float ist = 1.f / rB[0];

  float* __restrict__ oA = out_att + (size_t)row * N_;
  float* __restrict__ oO = out_orig + (size_t)row * N_;
  float* __restrict__ rg = (n < L_) ? (regular + ((size_t)b * L_ + n) * N_) : nullptr;
  #pragma unroll
  for (int p = 0; p < 3; ++p) {
    const int j = tid + p * 256;
    const float att = lt[j] * ist;
    oA[j] = att;
    oO[j] = le[j] * ise;
    if (rg) rg[j] = 0.5f * (float)arow[j] + 0.5f * att;
  }
}

// ---------------------------------------------------------------------------
// K7: h_block[b, l, e] = sum_{n'} regular[b, l, n'] * Wh[b, n', e]
// via V_WMMA_F32_16X16X4_F32; the result is tiled T_ times into out0.
// waves = B * (L/16) * (DOUT/16) = 32*16*8 = 4096
// ---------------------------------------------------------------------------
__global__ void __launch_bounds__(256) k_out_gemm(const float* __restrict__ regular,
                                                  const float* __restrict__ Wh,
                                                  float* __restrict__ out0) {
  const int wave = blockIdx.x * 8 + (threadIdx.x >> 5);
  const int lane = threadIdx.x & 31;
  const int nt = wave & 7;
  const int mt = (wave >> 3) & 15;
  const int b = wave >> 7;                // 0..31

  const float* __restrict__ A  = regular + (size_t)b * L_ * N_;   // [L, N]
  const float* __restrict__ Bm = Wh + (size_t)b * N_ * DOUT_;     // [N, DOUT]

  const int half = lane >> 4;
  const int lr   = lane & 15;
  const int m = mt * 16 + lr;
  const int n = nt * 16 + lr;

  v8f acc = {};
  #pragma unroll 4
  for (int k0 = 0; k0 < N_; k0 += 4) {
    const int kk = k0 + half * 2;
    v2f a, bb;
    a.x = A[(size_t)m * N_ + kk];
    a.y = A[(size_t)m * N_ + kk + 1];
    bb.x = Bm[(size_t)kk * DOUT_ + n];
    bb.y = Bm[(size_t)(kk + 1) * DOUT_ + n];
    acc = __builtin_amdgcn_wmma_f32_16x16x4_f32(false, a, false, bb,
                                                (short)0, acc, false, false);
  }
  #pragma unroll
  for (int v = 0; v < 8; ++v) {
    const int row = mt * 16 + half * 8 + v;
    const int col = nt * 16 + lr;
    const float val = acc[v];
    // faithful tiling: all T_ output blocks are identical
    #pragma unroll
    for (int t = 0; t < T_; ++t)
      out0[((size_t)b * N_ + t * L_ + row) * DOUT_ + col] = val;
  }
}

// ---------------------------------------------------------------------------
// Launcher
// ---------------------------------------------------------------------------
extern "C" void kernel_launch(void* const* d_in, const int* in_sizes, int n_in,
                              void* d_out, int out_size, void* d_ws, size_t ws_size,
                              hipStream_t stream) {
  const float* h      = (const float*)d_in[0];
  const int*   adj    = (const int*)d_in[1];
  const float* W      = (const float*)d_in[2];
  const float* a_node = (const float*)d_in[3];
  const float* a_type = (const float*)d_in[4];

  float* out = (float*)d_out;
  // output regions (flat, return order)
  float* out_h    = out;                                        // [B,N,DOUT]
  float* out_nta  = out + (size_t)B_ * N_ * DOUT_;              // [B,N,T]
  float* out_att  = out_nta + (size_t)B_ * N_ * T_;             // [B,N,N]
  float* out_orig = out_att + (size_t)B_ * N_ * N_;             // [B,N,N]

  // workspace regions (floats)
  float* ws      = (float*)d_ws;
  float* Wh      = ws;                                          // 3,145,728
  float* w1      = Wh + (size_t)B_ * N_ * DOUT_;                //    73,728
  float* w2      = w1 + (size_t)B_ * T_ * T_ * L_;              //    73,728
  float* e_type  = w2 + (size_t)B_ * T_ * T_ * L_;              //    73,728
  float* s1      = e_type + (size_t)B_ * T_ * T_ * L_;          //    24,576
  float* s2      = s1 + (size_t)B_ * N_;                        //    73,728
  float* regular = s2 + (size_t)B_ * T_ * N_;                   // 6,291,456

  // K1: Wh projection (WMMA f32)
  k_wh_gemm<<<(B_ * T_ * 16 * 8) / 8, 256, 0, stream>>>(h, W, Wh);

  // K2: per-node type-attention partials
  k_w12<<<(B_ * T_ * T_ * L_ + 255) / 256, 256, 0, stream>>>(Wh, a_type, w1, w2);

  // K3: adjacency-weighted e_type (replaces the msg GEMM algebraically)
  k_etype<<<(B_ * T_ * T_ * L_) / 8, 256, 0, stream>>>(adj, w1, w2, e_type);

  // K4: type softmax -> node_type_att_matrix output
  k_typeatt<<<(B_ * T_ * L_ + 255) / 256, 256, 0, stream>>>(e_type, out_nta);

  // K5: rank-1 node attention logits
  k_s12<<<(B_ * T_ * N_ + 255) / 256, 256, 0, stream>>>(Wh, a_node, s1, s2);

  // K6: masked double softmax rows -> attention, original_attention, regular
  k_attn_rows<<<B_ * N_, 256, 0, stream>>>(adj, s1, s2, out_nta,
                                           out_att, out_orig, regular);

  // K7: final GEMM (WMMA f32) + tile x3 into h_output
  k_out_gemm<<<(B_ * 16 * 8) / 8, 256, 0, stream>>>(regular, Wh, out_h);
}